// WordCharCNNEmbedding_60352880443609
// MI455X (gfx1250) — compile-verified
//
#include <hip/hip_runtime.h>

// ---------------------------------------------------------------------------
// WordCharCNNEmbedding for MI455X (gfx1250, wave32).
//
// char path : ce = char_emb_w[X]            [N,16,64]
//             conv1d(64->256,k=3,pad=2) -> relu -> maxpool_t -> [N,256]
// word path : word_emb_w[X_word]            [N,300]
// out       : concat                        [N,556] f32
//
// Conv decomposed per kernel tap k: P_k[s,o] = sum_c ce[s,c] * W_k[c,o]
// via v_wmma_f32_16x16x4_f32 (K accumulated over 16 chunks of 4).
// Tap results are scattered into Y[t][o] at t = s + 2 - k with ds_add_f32,
// then feat[o] = max(0, b[o] + max_t Y[t][o])   (relu folded into max).
//
// Key bandwidth decision: the packed conv weights (196 KB) are preloaded into
// VGPRs once per wave (96 VGPRs) and reused across 8 words per block,
// cutting weight re-fetch traffic from ~3.1 GB to ~384 MB.
// ---------------------------------------------------------------------------

typedef __attribute__((ext_vector_type(2))) float v2f;
typedef __attribute__((ext_vector_type(8))) float v8f;

#define N_WORDS  16384
#define L_CHARS  16
#define C_EMB    64
#define CONV_L   256
#define KERNEL_W 3
#define D_EMB    300
#define OUT_COLS (CONV_L + D_EMB)           // 556
#define N_OGRP   (CONV_L / 16)              // 16 output-channel groups
#define N_FRAG   (KERNEL_W * N_OGRP * 16)   // 3*16*16 = 768 B-fragments
#define WPB      8                          // words per block
#define T_LEN    (L_CHARS + 2)              // 18 conv output positions

// ---------------------------------------------------------------------------
// Kernel 1: repack conv_w (layout O=256, I=64, H=3) into exact WMMA
// B-fragment layout so the GEMM fragment loads are coalesced b64 loads.
// Fragment (k, g, q): B tile = W_k[4q..4q+3][g*16..g*16+15]  (K=4 x N=16 f32)
// B lane layout (mirror of ISA A 16x4 f32): lane L -> N = L%16,
//   word0: K = 4q + (L<16 ? 0 : 2), word1: K+1.
// ---------------------------------------------------------------------------
__global__ void pack_conv_w_kernel(const float* __restrict__ conv_w,
                                   float* __restrict__ wpack) {
    int tid  = blockIdx.x * blockDim.x + threadIdx.x;   // 0 .. 768*32-1
    if (tid >= N_FRAG * 32) return;
    int lane = tid & 31;
    int frag = tid >> 5;                // k*256 + g*16 + q
    int q    = frag & 15;
    int g    = (frag >> 4) & 15;
    int k    = frag >> 8;
    int koff = (lane < 16) ? 0 : 2;
    int o    = g * 16 + (lane & 15);
    int c0   = 4 * q + koff;
    float w0 = conv_w[o * (C_EMB * KERNEL_W) + (c0 + 0) * KERNEL_W + k];
    float w1 = conv_w[o * (C_EMB * KERNEL_W) + (c0 + 1) * KERNEL_W + k];
    wpack[frag * 64 + lane * 2 + 0] = w0;
    wpack[frag * 64 + lane * 2 + 1] = w1;
}

// ---------------------------------------------------------------------------
// Kernel 2: word-embedding gather (pure bandwidth): out[n, 256 + j]
// ---------------------------------------------------------------------------
__global__ void word_gather_kernel(const int* __restrict__ X_word,
                                   const float* __restrict__ word_emb,
                                   float* __restrict__ out) {
    long long i = (long long)blockIdx.x * blockDim.x + threadIdx.x;
    if (i >= (long long)N_WORDS * D_EMB) return;
    int n = (int)(i / D_EMB);
    int j = (int)(i % D_EMB);
    out[(long long)n * OUT_COLS + CONV_L + j] =
        word_emb[(long long)X_word[n] * D_EMB + j];
}

// ---------------------------------------------------------------------------
// Kernel 3: char conv. 512-thread block = 16 waves, one o-group each,
// processing WPB=8 words with register-resident B fragments.
// ---------------------------------------------------------------------------
__global__ __launch_bounds__(512)
void char_conv_kernel(const int* __restrict__ X,
                      const float* __restrict__ char_emb,
                      const float* __restrict__ wpack,
                      const float* __restrict__ conv_b,
                      float* __restrict__ out) {
    __shared__ int   Xs[WPB * L_CHARS];              // 128 ints
    __shared__ float ce[WPB * L_CHARS * C_EMB];      // [w][s][c], 32 KB
    __shared__ float Y[16 * T_LEN * 16];             // [wave][t][o], 18.4 KB

    const int tid = threadIdx.x;
    const int n0  = blockIdx.x * WPB;

    if (tid < WPB * L_CHARS) Xs[tid] = X[n0 * L_CHARS + tid];
    __syncthreads();

    // Gather char embeddings for 8 words: 8192 floats, 512 threads -> 16 each.
    for (int e = tid; e < WPB * L_CHARS * C_EMB; e += 512) {
        int ws = e >> 6;                 // w*16 + s
        int c  = e & 63;
        ce[e] = char_emb[Xs[ws] * C_EMB + c];
    }

    const int lane = tid & 31;
    const int wave = tid >> 5;          // == o-group g
    const int g    = wave;
    const int koff = (lane < 16) ? 0 : 2;   // A-fragment K offset per ISA layout
    const int srow = lane & 15;             // A-fragment row M
    const int shalf = (lane >> 4) << 3;     // C rows: M = v + shalf
    const int ocol  = lane & 15;            // C cols: N = lane%16

    // Preload all 48 B fragments for this wave's o-group into registers.
    v2f B0[16], B1[16], B2[16];
    {
        const v2f* wp = (const v2f*)wpack;
#pragma unroll
        for (int q = 0; q < 16; ++q) {
            B0[q] = wp[((0 * N_OGRP + g) * 16 + q) * 32 + lane];
            B1[q] = wp[((1 * N_OGRP + g) * 16 + q) * 32 + lane];
            B2[q] = wp[((2 * N_OGRP + g) * 16 + q) * 32 + lane];
        }
    }
    __syncthreads();                     // ce ready for all waves

    float* Yw = &Y[wave * (T_LEN * 16)];
    const float bias = conv_b[g * 16 + ocol];   // only lanes 0-15 use it

    for (int w = 0; w < WPB; ++w) {
        // Zero this wave's Y scatter buffer (wave-local, LDS is in-order).
#pragma unroll
        for (int i = lane; i < T_LEN * 16; i += 32) Yw[i] = 0.0f;

        const float* cew = &ce[w * L_CHARS * C_EMB];

        v8f acc0 = {}; v8f acc1 = {}; v8f acc2 = {};
#pragma unroll
        for (int q = 0; q < 16; ++q) {
            // A fragment: ce[w][s][4q+koff .. +1], 8B aligned
            v2f a = *(const v2f*)&cew[srow * C_EMB + 4 * q + koff];
            // 8-arg form: (neg_a, A, neg_b, B, c_mod, C, reuse_a, reuse_b)
            acc0 = __builtin_amdgcn_wmma_f32_16x16x4_f32(false, a, false, B0[q],
                                                         (short)0, acc0, false, false);
            acc1 = __builtin_amdgcn_wmma_f32_16x16x4_f32(false, a, false, B1[q],
                                                         (short)0, acc1, false, false);
            acc2 = __builtin_amdgcn_wmma_f32_16x16x4_f32(false, a, false, B2[q],
                                                         (short)0, acc2, false, false);
        }

        // Scatter-accumulate taps into Y[t][o] at t = s + 2 - k (always 0..17).
        // C layout: VGPR v -> M = v + shalf (= s), N = ocol.
#pragma unroll
        for (int v = 0; v < 8; ++v) {
            int s = v + shalf;
            atomicAdd(&Yw[(s + 2) * 16 + ocol], acc0[v]);   // k=0
            atomicAdd(&Yw[(s + 1) * 16 + ocol], acc1[v]);   // k=1
            atomicAdd(&Yw[(s + 0) * 16 + ocol], acc2[v]);   // k=2
        }
        __syncthreads();    // order LDS scatter before reduce (uniform loop)

        // Maxpool over t = 0..17, relu folded into the final max.
        if (lane < 16) {
            float smax = -3.402823466e38f;
#pragma unroll
            for (int t = 0; t < T_LEN; ++t)
                smax = fmaxf(smax, Yw[t * 16 + ocol]);
            float r = fmaxf(0.0f, bias + smax);
            out[(long long)(n0 + w) * OUT_COLS + g * 16 + ocol] = r;
        }
        __syncthreads();    // reduce reads done before next word's zeroing
    }
}

// ---------------------------------------------------------------------------
extern "C" void kernel_launch(void* const* d_in, const int* in_sizes, int n_in,
                              void* d_out, int out_size, void* d_ws, size_t ws_size,
                              hipStream_t stream) {
    const int*   X          = (const int*)d_in[0];
    const int*   X_word     = (const int*)d_in[1];
    const float* char_emb_w = (const float*)d_in[2];
    const float* conv_w     = (const float*)d_in[3];
    const float* conv_b     = (const float*)d_in[4];
    const float* word_emb_w = (const float*)d_in[5];
    float*       out        = (float*)d_out;
    float*       wpack      = (float*)d_ws;      // 768 frags * 64 f32 = 196608 B

    (void)in_sizes; (void)n_in; (void)out_size; (void)ws_size;

    // 1) Repack conv weights into WMMA B-fragment layout (L2-resident, 196 KB).
    pack_conv_w_kernel<<<(N_FRAG * 32 + 255) / 256, 256, 0, stream>>>(conv_w, wpack);

    // 2) Word-embedding gather (pure bandwidth path).
    long long wn = (long long)N_WORDS * D_EMB;
    word_gather_kernel<<<(int)((wn + 255) / 256), 256, 0, stream>>>(X_word, word_emb_w, out);

    // 3) Char conv via f32 WMMA: one block (16 waves) per 8 words.
    char_conv_kernel<<<N_WORDS / WPB, 512, 0, stream>>>(X, char_emb_w, wpack, conv_b, out);
}